// GraphNeuralNetwork_62105227100223
// MI455X (gfx1250) — compile-verified
//
#include <hip/hip_runtime.h>
#include <hip/hip_bf16.h>

#define N_NODES  100000
#define N_EDGES  3200000
#define N_FEAT   128
#define HIDDEN   64
#define N_GRAPHS 512
#define N_CLASSES 10

typedef __attribute__((ext_vector_type(2))) float v2f;
typedef __attribute__((ext_vector_type(8))) float v8f;

// ---------------------------------------------------------------------------
// Degree / norm kernels
// ---------------------------------------------------------------------------
__global__ void init_deg_kernel(float* __restrict__ deg, int n) {
    int i = blockIdx.x * blockDim.x + threadIdx.x;
    if (i < n) deg[i] = 1.0f;  // self-loop contributes 1 to every node's degree
}

__global__ void edge_deg_kernel(const int* __restrict__ dst, float* __restrict__ deg, int e) {
    int i = blockIdx.x * blockDim.x + threadIdx.x;
    if (i < e) atomicAdd(&deg[dst[i]], 1.0f);
}

__global__ void dinv_kernel(float* __restrict__ deg, int n) {
    int i = blockIdx.x * blockDim.x + threadIdx.x;
    if (i < n) {
        float d = deg[i];
        deg[i] = (d > 0.0f) ? rsqrtf(fmaxf(d, 1.0f)) : 0.0f;  // in-place: deg -> dinv
    }
}

// ---------------------------------------------------------------------------
// Strip-mined WMMA f32 GEMM: C[M,N] = A[M,K] @ B[K,N] (+bias, +relu optional)
// One wave computes one 16 x (16*NT) strip of C with V_WMMA_F32_16X16X4_F32,
// reusing a single A fragment across NT WMMAs per k-step (NT accumulators
// live in registers; NT=4 -> 32 accumulator VGPRs, well within budget).
// Requires M%16==0, N%(16*NT)==0, K%4==0.
// A 16x4 frag : lanes 0-15 hold (K=k0,k0+1) in v0,v1 ; lanes 16-31 hold (k0+2,k0+3)
// B 4x16 frag : v0 = row k0 (lanes 0-15) / row k0+2 (lanes 16-31); v1 = k0+1 / k0+3
// C/D         : vgpr r -> M = r (lanes 0-15) or r+8 (lanes 16-31), N = lane&15
// ---------------------------------------------------------------------------
template <int NT>
__global__ void gemm_f32_wmma_kernel(const float* __restrict__ A,
                                     const float* __restrict__ B,
                                     const float* __restrict__ bias,
                                     float* __restrict__ C,
                                     int M, int K, int N, int do_relu) {
    int wave = (int)((blockIdx.x * (long long)blockDim.x + threadIdx.x) >> 5);
    int lane = threadIdx.x & 31;
    int nstrips = N / (16 * NT);
    int m0  = (wave / nstrips) << 4;
    int ns0 = (wave % nstrips) * (16 * NT);
    if (m0 >= M) return;               // wave-uniform: EXEC stays all-ones for WMMA

    int half = lane >> 4;              // 0: lanes 0-15, 1: lanes 16-31
    int l    = lane & 15;

    v8f acc[NT];
#pragma unroll
    for (int t = 0; t < NT; ++t) acc[t] = (v8f){};

    const float* arow = A + (size_t)(m0 + l) * K + 2 * half;
    const float* brow = B + (size_t)(2 * half) * N + ns0 + l;
    for (int k0 = 0; k0 < K; k0 += 4) {
        v2f a;
        a.x = arow[k0];
        a.y = arow[k0 + 1];
#pragma unroll
        for (int t = 0; t < NT; ++t) {
            v2f b;
            b.x = brow[(size_t)k0 * N + 16 * t];
            b.y = brow[(size_t)(k0 + 1) * N + 16 * t];
            acc[t] = __builtin_amdgcn_wmma_f32_16x16x4_f32(
                /*neg_a=*/false, a, /*neg_b=*/false, b,
                /*c_mod=*/(short)0, acc[t], /*reuse_a=*/false, /*reuse_b=*/false);
        }
    }

#pragma unroll
    for (int t = 0; t < NT; ++t) {
        int ncol = ns0 + 16 * t + l;
        float bv = bias ? bias[ncol] : 0.0f;
#pragma unroll
        for (int r = 0; r < 8; ++r) {
            int m = m0 + r + 8 * half;
            float v = acc[t][r] + bv;
            if (do_relu) v = fmaxf(v, 0.0f);
            C[(size_t)m * N + ncol] = v;
        }
    }
}

// ---------------------------------------------------------------------------
// out[i][c] = dinv[i]^2 * h[i][c] + bias[c]   (self-loop message + bias)
// ---------------------------------------------------------------------------
__global__ void init_out_kernel(const float* __restrict__ h,
                                const float* __restrict__ dinv,
                                const float* __restrict__ bias,
                                float* __restrict__ out, int n) {
    long long t = blockIdx.x * (long long)blockDim.x + threadIdx.x;
    if (t >= (long long)n * HIDDEN) return;
    int i = (int)(t >> 6);       // HIDDEN == 64
    int c = (int)(t & 63);
    float di = dinv[i];
    out[t] = di * di * h[t] + bias[c];
}

// ---------------------------------------------------------------------------
// Edge scatter: out[dst] += dinv[src]*dinv[dst] * h[src]
// 16 threads per edge, each handles 4 consecutive floats (float4 gather,
// 4x global_atomic_add_f32 -- destination rows live in the 192MB L2).
// ---------------------------------------------------------------------------
__global__ void edge_scatter_kernel(const int* __restrict__ src,
                                    const int* __restrict__ dst,
                                    const float* __restrict__ dinv,
                                    const float* __restrict__ h,
                                    float* __restrict__ out, int e) {
    long long t = blockIdx.x * (long long)blockDim.x + threadIdx.x;
    int eidx = (int)(t >> 4);
    if (eidx >= e) return;
    int c = ((int)t & 15) << 2;
    int s = src[eidx];
    int d = dst[eidx];
    float norm = dinv[s] * dinv[d];
    const float4 v = *(const float4*)(h + (size_t)s * HIDDEN + c);
    float* op = out + (size_t)d * HIDDEN + c;
    atomicAdd(op + 0, norm * v.x);
    atomicAdd(op + 1, norm * v.y);
    atomicAdd(op + 2, norm * v.z);
    atomicAdd(op + 3, norm * v.w);
}

__global__ void relu_kernel(float* __restrict__ p, long long n) {
    long long t = blockIdx.x * (long long)blockDim.x + threadIdx.x;
    if (t < n) p[t] = fmaxf(p[t], 0.0f);
}

__global__ void zero_kernel(float* __restrict__ p, int n) {
    int t = blockIdx.x * blockDim.x + threadIdx.x;
    if (t < n) p[t] = 0.0f;
}

// ---------------------------------------------------------------------------
// Mean pool over batch ids
// ---------------------------------------------------------------------------
__global__ void pool_kernel(const float* __restrict__ h,
                            const int* __restrict__ batch,
                            float* __restrict__ pooled,
                            float* __restrict__ counts, int n) {
    long long t = blockIdx.x * (long long)blockDim.x + threadIdx.x;
    if (t >= (long long)n * HIDDEN) return;
    int i = (int)(t >> 6);
    int c = (int)(t & 63);
    int g = batch[i];
    atomicAdd(&pooled[(size_t)g * HIDDEN + c], h[t]);
    if (c == 0) atomicAdd(&counts[g], 1.0f);
}

__global__ void pool_div_kernel(float* __restrict__ pooled,
                                const float* __restrict__ counts) {
    int t = blockIdx.x * blockDim.x + threadIdx.x;
    if (t >= N_GRAPHS * HIDDEN) return;
    pooled[t] /= fmaxf(counts[t >> 6], 1.0f);
}

// ---------------------------------------------------------------------------
// Final tiny layer: out[g][c] = z[g] . Wc2[:,c] + bc2[c]   (512x32 @ 32x10)
// ---------------------------------------------------------------------------
__global__ void cls2_kernel(const float* __restrict__ z,
                            const float* __restrict__ Wc2,
                            const float* __restrict__ bc2,
                            float* __restrict__ out) {
    int t = blockIdx.x * blockDim.x + threadIdx.x;
    if (t >= N_GRAPHS * N_CLASSES) return;
    int g = t / N_CLASSES;
    int c = t % N_CLASSES;
    float acc = bc2[c];
#pragma unroll
    for (int k = 0; k < 32; ++k)
        acc = fmaf(z[g * 32 + k], Wc2[k * N_CLASSES + c], acc);
    out[t] = acc;
}

// ---------------------------------------------------------------------------
static inline int cdiv_ll(long long a, int b) { return (int)((a + b - 1) / b); }

extern "C" void kernel_launch(void* const* d_in, const int* in_sizes, int n_in,
                              void* d_out, int out_size, void* d_ws, size_t ws_size,
                              hipStream_t stream) {
    const float* x    = (const float*)d_in[0];
    const int*   edge = (const int*)d_in[1];
    const int*   batch= (const int*)d_in[2];
    const float* W1 = (const float*)d_in[3];  const float* b1  = (const float*)d_in[4];
    const float* W2 = (const float*)d_in[5];  const float* b2  = (const float*)d_in[6];
    const float* W3 = (const float*)d_in[7];  const float* b3  = (const float*)d_in[8];
    const float* Wc1= (const float*)d_in[9];  const float* bc1 = (const float*)d_in[10];
    const float* Wc2= (const float*)d_in[11]; const float* bc2 = (const float*)d_in[12];
    float* out = (float*)d_out;

    const int* src = edge;            // edge_index[0]
    const int* dst = edge + N_EDGES;  // edge_index[1]

    // Workspace layout (floats)
    float* dinv   = (float*)d_ws;                                   // N  (deg -> dinv)
    float* bufA   = dinv + ((N_NODES + 63) & ~63);                  // N*64  (GEMM out / msgs)
    float* bufB   = bufA + (size_t)N_NODES * HIDDEN;                // N*64  (scatter out / act)
    float* pooled = bufB + (size_t)N_NODES * HIDDEN;                // G*64
    float* counts = pooled + (size_t)N_GRAPHS * HIDDEN;             // G     (contiguous w/ pooled)
    float* zbuf   = counts + N_GRAPHS;                              // G*32

    const int B = 256;
    const long long NH = (long long)N_NODES * HIDDEN;

    // --- degree + symmetric norm ---
    init_deg_kernel<<<cdiv_ll(N_NODES, B), B, 0, stream>>>(dinv, N_NODES);
    edge_deg_kernel<<<cdiv_ll(N_EDGES, B), B, 0, stream>>>(dst, dinv, N_EDGES);
    dinv_kernel<<<cdiv_ll(N_NODES, B), B, 0, stream>>>(dinv, N_NODES);

    // --- GCN layer 1: K = 128 ---
    {
        long long waves = (long long)(N_NODES / 16) * (HIDDEN / 64);  // NT=4: one strip of 64
        gemm_f32_wmma_kernel<4><<<cdiv_ll(waves * 32, B), B, 0, stream>>>(
            x, W1, nullptr, bufA, N_NODES, N_FEAT, HIDDEN, 0);
        init_out_kernel<<<cdiv_ll(NH, B), B, 0, stream>>>(bufA, dinv, b1, bufB, N_NODES);
        edge_scatter_kernel<<<cdiv_ll((long long)N_EDGES * 16, B), B, 0, stream>>>(
            src, dst, dinv, bufA, bufB, N_EDGES);
        relu_kernel<<<cdiv_ll(NH, B), B, 0, stream>>>(bufB, NH);
    }
    // --- GCN layers 2,3: K = 64 ---
    const float* Ws[2] = {W2, W3};
    const float* bs[2] = {b2, b3};
    for (int layer = 0; layer < 2; ++layer) {
        long long waves = (long long)(N_NODES / 16) * (HIDDEN / 64);
        gemm_f32_wmma_kernel<4><<<cdiv_ll(waves * 32, B), B, 0, stream>>>(
            bufB, Ws[layer], nullptr, bufA, N_NODES, HIDDEN, HIDDEN, 0);
        init_out_kernel<<<cdiv_ll(NH, B), B, 0, stream>>>(bufA, dinv, bs[layer], bufB, N_NODES);
        edge_scatter_kernel<<<cdiv_ll((long long)N_EDGES * 16, B), B, 0, stream>>>(
            src, dst, dinv, bufA, bufB, N_EDGES);
        relu_kernel<<<cdiv_ll(NH, B), B, 0, stream>>>(bufB, NH);
    }

    // --- global mean pool ---
    zero_kernel<<<cdiv_ll(N_GRAPHS * HIDDEN + N_GRAPHS, B), B, 0, stream>>>(
        pooled, N_GRAPHS * HIDDEN + N_GRAPHS);
    pool_kernel<<<cdiv_ll(NH, B), B, 0, stream>>>(bufB, batch, pooled, counts, N_NODES);
    pool_div_kernel<<<cdiv_ll(N_GRAPHS * HIDDEN, B), B, 0, stream>>>(pooled, counts);

    // --- classifier: z = relu(pooled @ Wc1 + bc1), then out = z @ Wc2 + bc2 ---
    {
        long long waves = (long long)(N_GRAPHS / 16) * (32 / 32);     // NT=2: one strip of 32
        gemm_f32_wmma_kernel<2><<<cdiv_ll(waves * 32, B), B, 0, stream>>>(
            pooled, Wc1, bc1, zbuf, N_GRAPHS, HIDDEN, 32, 1);
    }
    cls2_kernel<<<cdiv_ll(N_GRAPHS * N_CLASSES, B), B, 0, stream>>>(zbuf, Wc2, bc2, out);
}